// AutoencoderGraphUnlearning_44057774522829
// MI455X (gfx1250) — compile-verified
//
#include <hip/hip_runtime.h>
#include <stdint.h>

#define N_NODES 150000
#define DFEAT   128
#define NBOT    64
#define NEDGE   4000000

typedef __attribute__((ext_vector_type(2))) float v2f;
typedef __attribute__((ext_vector_type(8))) float v8f;

__device__ __forceinline__ void atomic_add_f32(float* p, float v) {
    __hip_atomic_fetch_add(p, v, __ATOMIC_RELAXED, __HIP_MEMORY_SCOPE_AGENT);
}

// ---------------- SpMM: out[row[e]] += val[e] * x[col[e]]  (one wave32 per edge)
__global__ void spmm_kernel(const int* __restrict__ row, const int* __restrict__ col,
                            const float* __restrict__ val,
                            const float* __restrict__ x, float* __restrict__ out) {
    unsigned tid  = blockIdx.x * blockDim.x + threadIdx.x;
    unsigned e    = tid >> 5;
    unsigned lane = tid & 31;
    if (e >= NEDGE) return;
    int   r = row[e], c = col[e];
    float v = val[e];
    float4 xv = ((const float4*)(x + (size_t)c * DFEAT))[lane];
    float* o = out + (size_t)r * DFEAT + lane * 4;
    atomic_add_f32(o + 0, v * xv.x);
    atomic_add_f32(o + 1, v * xv.y);
    atomic_add_f32(o + 2, v * xv.z);
    atomic_add_f32(o + 3, v * xv.w);
}

__global__ void spmm_mask_kernel(const int* __restrict__ row, const int* __restrict__ col,
                                 const float* __restrict__ val,
                                 const uint8_t* __restrict__ mask, float invkeep,
                                 const float* __restrict__ x, float* __restrict__ out) {
    unsigned tid  = blockIdx.x * blockDim.x + threadIdx.x;
    unsigned e    = tid >> 5;
    unsigned lane = tid & 31;
    if (e >= NEDGE) return;
    if (!mask[e]) return;                       // wave-uniform branch
    int   r = row[e], c = col[e];
    float v = val[e] * invkeep;
    float4 xv = ((const float4*)(x + (size_t)c * DFEAT))[lane];
    float* o = out + (size_t)r * DFEAT + lane * 4;
    atomic_add_f32(o + 0, v * xv.x);
    atomic_add_f32(o + 1, v * xv.y);
    atomic_add_f32(o + 2, v * xv.z);
    atomic_add_f32(o + 3, v * xv.w);
}

// ---------------- Dense GEMM via V_WMMA_F32_16X16X4_F32
// Block = 256 threads (8 waves). W (K x NC, <=32KB) staged in LDS once per block.
// Each wave owns one 16-row M-tile and accumulates all TN = NC/16 column tiles,
// so each A b64 fragment load feeds TN WMMAs; B fragments come from LDS.
template <int K, int NC>
__global__ __launch_bounds__(256) void wmma_gemm_lds_kernel(
        const float* __restrict__ A, const float* __restrict__ W,
        const float* __restrict__ bias, float* __restrict__ O) {
    constexpr int TN = NC / 16;
    __shared__ float sW[K * NC];                 // 32 KB for both instantiations

    // cooperative weight staging (float4 = global_load_b128)
    for (int i = threadIdx.x; i < (K * NC) / 4; i += 256)
        ((float4*)sW)[i] = ((const float4*)W)[i];
    __syncthreads();

    unsigned waveInBlk = threadIdx.x >> 5;
    unsigned lane      = threadIdx.x & 31;
    unsigned tm        = blockIdx.x * 8 + waveInBlk;
    if (tm >= N_NODES / 16) return;              // wave-uniform, after the barrier

    int half = lane >> 4;                        // 0: K pair {0,1}; 1: K pair {2,3}
    int l15  = lane & 15;

    const float* arow = A + (size_t)(tm * 16 + l15) * K + half * 2;

    v8f acc[TN];
#pragma unroll
    for (int t = 0; t < TN; t++) acc[t] = {};

    for (int k = 0; k < K; k += 4) {
        v2f a;
        a.x = arow[k];
        a.y = arow[k + 1];
        const float* w0 = sW + (size_t)(k + half * 2) * NC + l15;
        const float* w1 = w0 + NC;
#pragma unroll
        for (int t = 0; t < TN; t++) {
            v2f b;
            b.x = w0[t * 16];
            b.y = w1[t * 16];
            acc[t] = __builtin_amdgcn_wmma_f32_16x16x4_f32(false, a, false, b,
                                                           (short)0, acc[t], false, false);
        }
    }

#pragma unroll
    for (int t = 0; t < TN; t++) {
        float  bv    = bias[t * 16 + l15];
        float* obase = O + (size_t)(tm * 16 + half * 8) * NC + t * 16 + l15;
#pragma unroll
        for (int i = 0; i < 8; i++) obase[(size_t)i * NC] = acc[t][i] + bv;
    }
}

// ---------------- elementwise helpers
__global__ void scale_wr_kernel(const float* __restrict__ fnl, const float* __restrict__ wr,
                                float* __restrict__ out, size_t n) {
    size_t i = blockIdx.x * (size_t)blockDim.x + threadIdx.x;
    if (i >= n) return;
    out[i] = fnl[i] * wr[i / DFEAT];
}

__global__ void sub_inplace_kernel(float* __restrict__ d, const float* __restrict__ s, size_t n) {
    size_t i = blockIdx.x * (size_t)blockDim.x + threadIdx.x;
    if (i >= n) return;
    d[i] -= s[i];
}

__global__ void leaky_inplace_kernel(float* __restrict__ d, size_t n) {
    size_t i = blockIdx.x * (size_t)blockDim.x + threadIdx.x;
    if (i >= n) return;
    float x = d[i];
    d[i] = x >= 0.0f ? x : 0.5f * x;
}

__global__ void add3_kernel(const float* __restrict__ a, const float* __restrict__ b,
                            const float* __restrict__ c, float* __restrict__ o, size_t n) {
    size_t i = blockIdx.x * (size_t)blockDim.x + threadIdx.x;
    if (i >= n) return;
    o[i] = a[i] + b[i] + c[i];
}

__global__ void final_kernel(const float* __restrict__ ini, const float* __restrict__ w,
                             const float* __restrict__ e, const float* __restrict__ zp,
                             float* __restrict__ o, size_t n) {
    size_t i = blockIdx.x * (size_t)blockDim.x + threadIdx.x;
    if (i >= n) return;
    float x = -0.5f * w[i] + e[i] + zp[i];
    x = x >= 0.0f ? x : 0.5f * x;   // LAYER_MLP leaky #1
    x = x >= 0.0f ? x : 0.5f * x;   // LAYER_MLP leaky #2
    o[i] = ini[i] + x;
}

extern "C" void kernel_launch(void* const* d_in, const int* in_sizes, int n_in,
                              void* d_out, int out_size, void* d_ws, size_t ws_size,
                              hipStream_t stream) {
    const int*     drp_row    = (const int*)d_in[0];
    const int*     drp_col    = (const int*)d_in[1];
    const float*   drp_val    = (const float*)d_in[2];
    const int*     pk_row     = (const int*)d_in[3];
    const int*     pk_col     = (const int*)d_in[4];
    const float*   pk_val     = (const float*)d_in[5];
    const uint8_t* hyper_mask = (const uint8_t*)d_in[6];
    const float*   edge1      = (const float*)d_in[7];
    const float*   ini        = (const float*)d_in[8];
    const float*   fnl        = (const float*)d_in[9];
    const float*   wr         = (const float*)d_in[10];
    const float*   enc_W      = (const float*)d_in[11];
    const float*   enc_b      = (const float*)d_in[12];
    const float*   dec_W      = (const float*)d_in[13];
    const float*   dec_b      = (const float*)d_in[14];

    const size_t ND = (size_t)N_NODES * DFEAT;
    float* out     = (float*)d_out;
    float* o_tuned = out;
    float* o_g0    = out + ND;
    float* o_g1    = out + 2 * ND;
    float* o_h0    = out + 3 * ND;
    float* o_h1    = out + 4 * ND;

    float* A  = (float*)d_ws;      // N x D
    float* B  = A + ND;            // N x D
    float* C  = B + ND;            // N x D
    float* Dd = C + ND;            // N x D
    float* Z  = Dd + ND;           // N x 64

    const dim3 blk(256);
    const unsigned spmm_blocks = (NEDGE * 32u + 255u) / 256u;     // 500000
    const unsigned ew_blocks   = (unsigned)((ND + 255) / 256);    // 75000
    const unsigned gemm_blocks = (N_NODES / 16 + 7) / 8;          // 1172

#define ZERO(p) hipMemsetAsync((p), 0, ND * sizeof(float), stream)
#define SPMM(rp, cp, vp, src, dst)                                                    \
    do { ZERO(dst);                                                                   \
         spmm_kernel<<<spmm_blocks, blk, 0, stream>>>((rp), (cp), (vp), (src), (dst)); } while (0)
#define SPMM_MASK(mp, src, dst)                                                       \
    do { ZERO(dst);                                                                   \
         spmm_mask_kernel<<<spmm_blocks, blk, 0, stream>>>(drp_row, drp_col, drp_val, \
                                                           (mp), 1.0f / 0.95f, (src), (dst)); } while (0)

    // --- GNN layers: tem/hyp written straight into output slices
    SPMM(drp_row, drp_col, drp_val, edge1, o_g0);
    SPMM_MASK(hyper_mask + 0,      edge1, o_h0);
    SPMM(drp_row, drp_col, drp_val, o_g0,  o_g1);
    SPMM_MASK(hyper_mask + NEDGE,  o_g0,   o_h1);

    // --- withdrawal propagation: w = drp^2 (fnl * wr)       -> B
    scale_wr_kernel<<<ew_blocks, blk, 0, stream>>>(fnl, wr, B, ND);
    SPMM(drp_row, drp_col, drp_val, B, C);
    SPMM(drp_row, drp_col, drp_val, C, B);

    // --- influence: H = drp^2 ini ; Ew = pk^2 ini ; inf = H - Ew   -> Dd
    SPMM(drp_row, drp_col, drp_val, ini, C);
    SPMM(drp_row, drp_col, drp_val, C, Dd);
    SPMM(pk_row, pk_col, pk_val, ini, C);
    SPMM(pk_row, pk_col, pk_val, C, A);
    sub_inplace_kernel<<<ew_blocks, blk, 0, stream>>>(Dd, A, ND);

    // --- GCN autoencoder body: h2 = leaky(drp leaky(drp inf))      -> Dd
    SPMM(drp_row, drp_col, drp_val, Dd, C);
    leaky_inplace_kernel<<<ew_blocks, blk, 0, stream>>>(C, ND);
    SPMM(drp_row, drp_col, drp_val, C, Dd);
    leaky_inplace_kernel<<<ew_blocks, blk, 0, stream>>>(Dd, ND);

    // --- WMMA encode (N,128)x(128,64) -> Z ; decode (N,64)x(64,128) -> C
    wmma_gemm_lds_kernel<DFEAT, NBOT><<<gemm_blocks, blk, 0, stream>>>(Dd, enc_W, enc_b, Z);
    wmma_gemm_lds_kernel<NBOT, DFEAT><<<gemm_blocks, blk, 0, stream>>>(Z, dec_W, dec_b, C);

    // --- edge propagation: e = pk^2 (edge1 + g0 + g1)              -> A
    add3_kernel<<<ew_blocks, blk, 0, stream>>>(edge1, o_g0, o_g1, A, ND);
    SPMM(pk_row, pk_col, pk_val, A, Dd);
    SPMM(pk_row, pk_col, pk_val, Dd, A);

    // --- tuned = ini + leaky(leaky(-0.5*w + e + Zp))
    final_kernel<<<ew_blocks, blk, 0, stream>>>(ini, B, A, C, o_tuned, ND);

#undef SPMM_MASK
#undef SPMM
#undef ZERO
}